// DisplaceChannel_45380624450258
// MI455X (gfx1250) — compile-verified
//
#include <hip/hip_runtime.h>

namespace {
constexpr int B = 16, C = 256, H = 128, W = 128, G = 32;
constexpr int BIND = C / G;            // 8 channels per group
constexpr int ROWS = 32;               // output rows per block (within one channel)
constexpr int SROWS = ROWS + 1;        // staged source rows (vertical halo)
constexpr int LDSW = 132;              // floats per staged row; 132*4B = 16B multiple
constexpr float OFFSET_SCALE = 128.0f;
constexpr int TILES_PER_BG = (BIND * H) / ROWS;   // 32
}

typedef float v4f __attribute__((ext_vector_type(4)));

#if __has_builtin(__builtin_amdgcn_global_load_async_to_lds_b128)
#define HAS_ASYNC_LDS 1
typedef int v4i __attribute__((vector_size(16)));
typedef __attribute__((address_space(1))) v4i* gv4i_ptr;   // global int4*
typedef __attribute__((address_space(3))) v4i* lv4i_ptr;   // LDS int4*
#else
#define HAS_ASYNC_LDS 0
#endif

__global__ __launch_bounds__(256) void displace_kernel(
    const float* __restrict__ inp, const float* __restrict__ off,
    float* __restrict__ out)
{
    __shared__ float smem[SROWS * LDSW];

    const int t = threadIdx.x;
    int idx = blockIdx.x;
    const int tile = idx % TILES_PER_BG; idx /= TILES_PER_BG;
    const int g    = idx % G;            idx /= G;
    const int b    = idx;

    // Group-uniform shift parameters
    const float dxv = off[2 * g + 0] * OFFSET_SCALE;
    const float dyv = off[2 * g + 1] * OFFSET_SCALE;
    const float xf = floorf(dxv), yf = floorf(dyv);
    const float fx = dxv - xf,  fy = dyv - yf;
    const int ix = (int)xf, iy = (int)yf;

    const int row0 = tile * ROWS;        // row index within (BIND*H)
    const int cl   = row0 / H;           // channel within group (tile never crosses)
    const int y0   = row0 % H;
    const int c    = g * BIND + cl;
    const size_t plane = ((size_t)(b * C + c)) * (size_t)(H * W);

    // ---- Stage SROWS clamped source rows into LDS (16B chunks, async DMA) ----
    for (int cidx = t; cidx < SROWS * (W / 4); cidx += 256) {
        const int k  = cidx >> 5;            // staged row (32 chunks per row)
        const int x4 = (cidx & 31) << 2;     // float offset within row
        int sy = y0 + iy + k;
        sy = sy < 0 ? 0 : (sy > H - 1 ? H - 1 : sy);   // clamp; masked at compute
        const float* gsrc = inp + plane + (size_t)sy * W + x4;
        float* ldst = &smem[k * LDSW + x4];
#if HAS_ASYNC_LDS
        __builtin_amdgcn_global_load_async_to_lds_b128(
            (gv4i_ptr)gsrc, (lv4i_ptr)ldst, 0, 0);
#else
        *reinterpret_cast<v4f*>(ldst) = *reinterpret_cast<const v4f*>(gsrc);
#endif
    }
#if HAS_ASYNC_LDS
    asm volatile("s_wait_asynccnt 0" ::: "memory");
#endif
    __syncthreads();

    // ---- Bilinear blend with OOB masks folded into weights ----
    const int lane = t & 31;
    const int x0 = lane << 2;
    const float omfx = 1.0f - fx, omfy = 1.0f - fy;

    for (int rr = (t >> 5); rr < ROWS; rr += 8) {
        const int y   = y0 + rr;
        const int sy0 = y + iy;
        const float v0 = (sy0 >= 0     && sy0 < H    ) ? 1.0f : 0.0f;
        const float v1 = (sy0 + 1 >= 0 && sy0 + 1 < H) ? 1.0f : 0.0f;
        const float w00 = omfy * omfx * v0;
        const float w01 = omfy * fx   * v0;
        const float w10 = fy   * omfx * v1;
        const float w11 = fy   * fx   * v1;
        const float* r0 = &smem[rr * LDSW];
        const float* r1 = r0 + LDSW;
        v4f res;
#pragma unroll
        for (int j = 0; j < 4; ++j) {
            const int sx0 = x0 + j + ix;
            const int sx1 = sx0 + 1;
            const float h0 = (sx0 >= 0 && sx0 < W) ? 1.0f : 0.0f;
            const float h1 = (sx1 >= 0 && sx1 < W) ? 1.0f : 0.0f;
            const int cx0 = sx0 < 0 ? 0 : (sx0 > W - 1 ? W - 1 : sx0);
            const int cx1 = sx1 < 0 ? 0 : (sx1 > W - 1 ? W - 1 : sx1);
            res[j] = w00 * (h0 * r0[cx0]) + w01 * (h1 * r0[cx1])
                   + w10 * (h0 * r1[cx0]) + w11 * (h1 * r1[cx1]);
        }
        // Output is write-once: non-temporal store keeps L2 free for input halo reuse.
        __builtin_nontemporal_store(res,
            reinterpret_cast<v4f*>(out + plane + (size_t)y * W + x0));
    }
}

extern "C" void kernel_launch(void* const* d_in, const int* in_sizes, int n_in,
                              void* d_out, int out_size, void* d_ws, size_t ws_size,
                              hipStream_t stream) {
    const float* inp = (const float*)d_in[0];   // (16,256,128,128) f32
    const float* off = (const float*)d_in[1];   // (32,2) f32
    float* out = (float*)d_out;                 // (16,256,128,128) f32

    const int nblocks = B * G * TILES_PER_BG;   // 16384
    displace_kernel<<<nblocks, 256, 0, stream>>>(inp, off, out);
}